// DenseFPModule_14482629722282
// MI455X (gfx1250) — compile-verified
//
#include <hip/hip_runtime.h>

// Problem constants (match reference)
#define B_    8
#define N0_   8192
#define N_    2048
#define CP_   512
#define CS_   256
#define CIN1_ (CP_ + CS_)   // 768
#define C1_   256
#define C2_   256
#define BN_EPS_ 1e-3f
#define SLOPE_  0.01f

typedef __attribute__((ext_vector_type(2))) float v2f;
typedef __attribute__((ext_vector_type(8))) float v8f;

// ---------------------------------------------------------------------------
// 0) passthrough copy of xyz into the first output slot
// ---------------------------------------------------------------------------
__global__ void copy_f32_kernel(const float* __restrict__ in,
                                float* __restrict__ out, int n) {
  int i = blockIdx.x * blockDim.x + threadIdx.x;
  if (i < n) out[i] = in[i];
}

// ---------------------------------------------------------------------------
// 1) 3-NN search: all 2048 source points cached in LDS, one query per thread
// ---------------------------------------------------------------------------
__global__ __launch_bounds__(256)
void knn3_kernel(const float* __restrict__ xyz,        // [B,3,N0]
                 const float* __restrict__ xyz_prev,   // [B,3,N]
                 int* __restrict__ idx3,               // [B,N0,3]
                 float* __restrict__ w3) {             // [B,N0,3]
  __shared__ float xs[N_], ys[N_], zs[N_];             // 24 KB LDS
  const int b = blockIdx.y;
  const int n = blockIdx.x * blockDim.x + threadIdx.x;
  const float* xp = xyz_prev + (size_t)b * 3 * N_;
  for (int j = threadIdx.x; j < N_; j += blockDim.x) {
    xs[j] = xp[j];
    ys[j] = xp[N_ + j];
    zs[j] = xp[2 * N_ + j];
  }
  __syncthreads();

  const float* q = xyz + (size_t)b * 3 * N0_;
  const float qx = q[n], qy = q[N0_ + n], qz = q[2 * N0_ + n];

  float d0 = 3.4e38f, d1 = 3.4e38f, d2 = 3.4e38f;
  int   i0 = 0, i1 = 0, i2 = 0;
  for (int j = 0; j < N_; ++j) {
    float dx = qx - xs[j], dy = qy - ys[j], dz = qz - zs[j];
    float d = dx * dx + dy * dy + dz * dz;
    if (d < d0)      { d2 = d1; i2 = i1; d1 = d0; i1 = i0; d0 = d; i0 = j; }
    else if (d < d1) { d2 = d1; i2 = i1; d1 = d;  i1 = j; }
    else if (d < d2) { d2 = d;  i2 = j; }
  }
  float v0 = 1.0f / (fmaxf(d0, 1e-10f) + 1e-8f);
  float v1 = 1.0f / (fmaxf(d1, 1e-10f) + 1e-8f);
  float v2 = 1.0f / (fmaxf(d2, 1e-10f) + 1e-8f);
  float inv = 1.0f / (v0 + v1 + v2);
  size_t o = ((size_t)b * N0_ + n) * 3;
  idx3[o] = i0; idx3[o + 1] = i1; idx3[o + 2] = i2;
  w3[o] = v0 * inv; w3[o + 1] = v1 * inv; w3[o + 2] = v2 * inv;
}

// ---------------------------------------------------------------------------
// 2) weighted gather: nfbuf[b,c,n] = sum_k w[b,n,k] * feat_prev[b,c,idx[b,n,k]]
//    feat_prev (33MB) is L2-resident -> gathers are cheap
// ---------------------------------------------------------------------------
__global__ __launch_bounds__(256)
void interp_kernel(const float* __restrict__ feat_prev,  // [B,CP,N]
                   const int* __restrict__ idx3,
                   const float* __restrict__ w3,
                   float* __restrict__ nfbuf) {          // [B,CP,N0]
  size_t i = (size_t)blockIdx.x * blockDim.x + threadIdx.x;
  int n = (int)(i % N0_);
  int c = (int)((i / N0_) % CP_);
  int b = (int)(i / ((size_t)N0_ * CP_));
  size_t o = ((size_t)b * N0_ + n) * 3;
  const float* fp = feat_prev + ((size_t)b * CP_ + c) * N_;
  float v = w3[o]     * fp[idx3[o]]
          + w3[o + 1] * fp[idx3[o + 1]]
          + w3[o + 2] * fp[idx3[o + 2]];
  nfbuf[i] = v;
}

// ---------------------------------------------------------------------------
// 3) GEMM1: Y1[b,o,n] = sum_c W1[o,c] * X[b,c,n], X = [nfbuf ; skip] along c.
//    fp32 WMMA 16x16x4. One block covers all 256 out-channels (8 waves),
//    each wave owns a 32(o) x 64(n) tile -> 8 v8f accumulators, so each
//    B fragment feeds 2 WMMAs (1.5 loaded dwords per WMMA).
// ---------------------------------------------------------------------------
__global__ __launch_bounds__(256)
void gemm1_wmma_kernel(const float* __restrict__ W1,     // [C1, 768]
                       const float* __restrict__ nfbuf,  // [B,CP,N0]
                       const float* __restrict__ skip,   // [B,CS,N0]
                       float* __restrict__ Y1) {         // [B,C1,N0]
  const int bz   = blockIdx.z;
  const int tid  = threadIdx.x;
  const int wave = tid >> 5;
  const int lane = tid & 31;
  const int hf   = lane >> 4;    // half-wave select (K pairs 0,1 vs 2,3)
  const int lid  = lane & 15;
  const int o0   = wave * 32;    // 8 waves x 32 = 256 = C1
  const int n0   = blockIdx.y * 64;
  const int nL   = n0 + lid;

  const float* Arow0 = W1 + (size_t)(o0 + lid) * CIN1_;
  const float* Arow1 = W1 + (size_t)(o0 + 16 + lid) * CIN1_;
  const float* Xa    = nfbuf + (size_t)bz * CP_ * N0_;
  const float* Xb    = skip  + (size_t)bz * CS_ * N0_;

  v8f acc[8] = {};

  // K part 1: interpolated features (c in [0,512))
  for (int k = 0; k < CP_; k += 4) {
    const int kk = k + 2 * hf;
    v2f a0; a0.x = Arow0[kk]; a0.y = Arow0[kk + 1];
    v2f a1; a1.x = Arow1[kk]; a1.y = Arow1[kk + 1];
    const float* xr = Xa + (size_t)kk * N0_;
    __builtin_prefetch(xr + 4 * N0_, 0, 0);
#pragma unroll
    for (int t = 0; t < 4; ++t) {
      v2f bf; bf.x = xr[nL + 16 * t]; bf.y = xr[N0_ + nL + 16 * t];
      acc[t] = __builtin_amdgcn_wmma_f32_16x16x4_f32(
          false, a0, false, bf, (short)0, acc[t], false, false);
      acc[4 + t] = __builtin_amdgcn_wmma_f32_16x16x4_f32(
          false, a1, false, bf, (short)0, acc[4 + t], false, false);
    }
  }
  // K part 2: skip features (c in [512,768))
  for (int k = 0; k < CS_; k += 4) {
    const int kk = k + 2 * hf;
    v2f a0; a0.x = Arow0[CP_ + kk]; a0.y = Arow0[CP_ + kk + 1];
    v2f a1; a1.x = Arow1[CP_ + kk]; a1.y = Arow1[CP_ + kk + 1];
    const float* xr = Xb + (size_t)kk * N0_;
    __builtin_prefetch(xr + 4 * N0_, 0, 0);
#pragma unroll
    for (int t = 0; t < 4; ++t) {
      v2f bf; bf.x = xr[nL + 16 * t]; bf.y = xr[N0_ + nL + 16 * t];
      acc[t] = __builtin_amdgcn_wmma_f32_16x16x4_f32(
          false, a0, false, bf, (short)0, acc[t], false, false);
      acc[4 + t] = __builtin_amdgcn_wmma_f32_16x16x4_f32(
          false, a1, false, bf, (short)0, acc[4 + t], false, false);
    }
  }

  float* Yb = Y1 + (size_t)bz * C1_ * N0_;
#pragma unroll
  for (int t = 0; t < 4; ++t)
#pragma unroll
    for (int r = 0; r < 8; ++r) {
      Yb[(size_t)(o0 + r + 8 * hf) * N0_ + n0 + 16 * t + lid]      = acc[t][r];
      Yb[(size_t)(o0 + 16 + r + 8 * hf) * N0_ + n0 + 16 * t + lid] = acc[4 + t][r];
    }
}

// ---------------------------------------------------------------------------
// 4) BatchNorm batch stats (train mode, biased var) -> fused scale/shift
//    one workgroup per channel, reduce 65536 values
// ---------------------------------------------------------------------------
__global__ __launch_bounds__(256)
void bn_stats_kernel(const float* __restrict__ Y, int C,
                     const float* __restrict__ g, const float* __restrict__ bb,
                     float* __restrict__ scale, float* __restrict__ shift) {
  __shared__ float ssum[256], ssq[256];
  const int o = blockIdx.x;
  float s = 0.0f, q = 0.0f;
  for (int i = threadIdx.x; i < B_ * N0_; i += 256) {
    int bidx = i / N0_, n = i - bidx * N0_;
    float v = Y[((size_t)bidx * C + o) * N0_ + n];
    s += v; q += v * v;
  }
  ssum[threadIdx.x] = s; ssq[threadIdx.x] = q;
  __syncthreads();
  for (int st = 128; st > 0; st >>= 1) {
    if (threadIdx.x < st) {
      ssum[threadIdx.x] += ssum[threadIdx.x + st];
      ssq[threadIdx.x]  += ssq[threadIdx.x + st];
    }
    __syncthreads();
  }
  if (threadIdx.x == 0) {
    const float invn = 1.0f / (float)(B_ * N0_);
    float mean = ssum[0] * invn;
    float var  = ssq[0] * invn - mean * mean;
    float sc   = g[o] * rsqrtf(var + BN_EPS_);
    scale[o] = sc;
    shift[o] = bb[o] - mean * sc;
  }
}

// ---------------------------------------------------------------------------
// 5) GEMM2: applies BN1 + LeakyReLU on the fly while loading B operand,
//    writes raw Y2 straight into d_out's nf region. Same 32x64 wave tile.
// ---------------------------------------------------------------------------
__global__ __launch_bounds__(256)
void gemm2_wmma_kernel(const float* __restrict__ W2,      // [C2, C1]
                       const float* __restrict__ Y1,      // [B,C1,N0] raw
                       const float* __restrict__ scale1,
                       const float* __restrict__ shift1,
                       float* __restrict__ Y2) {          // [B,C2,N0]
  const int bz   = blockIdx.z;
  const int tid  = threadIdx.x;
  const int wave = tid >> 5;
  const int lane = tid & 31;
  const int hf   = lane >> 4;
  const int lid  = lane & 15;
  const int o0   = wave * 32;    // 8 waves x 32 = 256 = C2
  const int n0   = blockIdx.y * 64;
  const int nL   = n0 + lid;

  const float* Arow0 = W2 + (size_t)(o0 + lid) * C1_;
  const float* Arow1 = W2 + (size_t)(o0 + 16 + lid) * C1_;
  const float* Xa    = Y1 + (size_t)bz * C1_ * N0_;

  v8f acc[8] = {};

  for (int k = 0; k < C1_; k += 4) {
    const int kk = k + 2 * hf;
    v2f a0; a0.x = Arow0[kk]; a0.y = Arow0[kk + 1];
    v2f a1; a1.x = Arow1[kk]; a1.y = Arow1[kk + 1];
    const float s0 = scale1[kk],     h0 = shift1[kk];
    const float s1 = scale1[kk + 1], h1 = shift1[kk + 1];
    const float* xr = Xa + (size_t)kk * N0_;
    __builtin_prefetch(xr + 4 * N0_, 0, 0);
#pragma unroll
    for (int t = 0; t < 4; ++t) {
      float u0 = s0 * xr[nL + 16 * t] + h0;
      float u1 = s1 * xr[N0_ + nL + 16 * t] + h1;
      u0 = (u0 >= 0.0f) ? u0 : SLOPE_ * u0;
      u1 = (u1 >= 0.0f) ? u1 : SLOPE_ * u1;
      v2f bf; bf.x = u0; bf.y = u1;
      acc[t] = __builtin_amdgcn_wmma_f32_16x16x4_f32(
          false, a0, false, bf, (short)0, acc[t], false, false);
      acc[4 + t] = __builtin_amdgcn_wmma_f32_16x16x4_f32(
          false, a1, false, bf, (short)0, acc[4 + t], false, false);
    }
  }

  float* Yb = Y2 + (size_t)bz * C2_ * N0_;
#pragma unroll
  for (int t = 0; t < 4; ++t)
#pragma unroll
    for (int r = 0; r < 8; ++r) {
      Yb[(size_t)(o0 + r + 8 * hf) * N0_ + n0 + 16 * t + lid]      = acc[t][r];
      Yb[(size_t)(o0 + 16 + r + 8 * hf) * N0_ + n0 + 16 * t + lid] = acc[4 + t][r];
    }
}

// ---------------------------------------------------------------------------
// 6) in-place BN2 + LeakyReLU on d_out nf region
// ---------------------------------------------------------------------------
__global__ __launch_bounds__(256)
void bn_act_kernel(float* __restrict__ Y,
                   const float* __restrict__ scale,
                   const float* __restrict__ shift, int C) {
  size_t i = (size_t)blockIdx.x * blockDim.x + threadIdx.x;
  int o = (int)((i / N0_) % C);
  float v = Y[i];
  v = scale[o] * v + shift[o];
  Y[i] = (v >= 0.0f) ? v : SLOPE_ * v;
}

// ---------------------------------------------------------------------------
// launcher
// ---------------------------------------------------------------------------
extern "C" void kernel_launch(void* const* d_in, const int* in_sizes, int n_in,
                              void* d_out, int out_size, void* d_ws, size_t ws_size,
                              hipStream_t stream) {
  (void)in_sizes; (void)n_in; (void)out_size; (void)ws_size;
  const float* xyz       = (const float*)d_in[0];  // [8,3,8192]
  const float* skip      = (const float*)d_in[1];  // [8,256,8192]
  const float* xyz_prev  = (const float*)d_in[2];  // [8,3,2048]
  const float* feat_prev = (const float*)d_in[3];  // [8,512,2048]
  const float* W1        = (const float*)d_in[4];  // [256,768]
  const float* g1        = (const float*)d_in[5];
  const float* b1        = (const float*)d_in[6];
  const float* W2        = (const float*)d_in[7];  // [256,256]
  const float* g2        = (const float*)d_in[8];
  const float* b2        = (const float*)d_in[9];

  float* out_xyz = (float*)d_out;                    // [8,3,8192]
  float* out_nf  = out_xyz + (size_t)B_ * 3 * N0_;   // [8,256,8192]

  // workspace carve-up (~203 MB)
  float* wsf    = (float*)d_ws;
  float* nfbuf  = wsf;                                   // B*CP*N0
  float* Y1     = nfbuf + (size_t)B_ * CP_ * N0_;        // B*C1*N0
  float* w3     = Y1 + (size_t)B_ * C1_ * N0_;           // B*N0*3
  float* scale1 = w3 + (size_t)B_ * N0_ * 3;
  float* shift1 = scale1 + C1_;
  float* scale2 = shift1 + C1_;
  float* shift2 = scale2 + C2_;
  int*   idx3   = (int*)(shift2 + C2_);                  // B*N0*3

  // 0) xyz passthrough
  copy_f32_kernel<<<(B_ * 3 * N0_) / 256, 256, 0, stream>>>(
      xyz, out_xyz, B_ * 3 * N0_);

  // 1) 3-NN + inverse-distance weights
  knn3_kernel<<<dim3(N0_ / 256, B_), 256, 0, stream>>>(xyz, xyz_prev, idx3, w3);

  // 2) weighted gather
  interp_kernel<<<(int)(((size_t)B_ * CP_ * N0_) / 256), 256, 0, stream>>>(
      feat_prev, idx3, w3, nfbuf);

  // 3) GEMM1 (concat handled by K-split); one block spans all 256 out-chans
  gemm1_wmma_kernel<<<dim3(1, N0_ / 64, B_), 256, 0, stream>>>(
      W1, nfbuf, skip, Y1);

  // 4) BN1 stats
  bn_stats_kernel<<<C1_, 256, 0, stream>>>(Y1, C1_, g1, b1, scale1, shift1);

  // 5) GEMM2 with fused BN1+LeakyReLU on input; raw Y2 into d_out
  gemm2_wmma_kernel<<<dim3(1, N0_ / 64, B_), 256, 0, stream>>>(
      W2, Y1, scale1, shift1, out_nf);

  // 6) BN2 stats over raw Y2
  bn_stats_kernel<<<C2_, 256, 0, stream>>>(out_nf, C2_, g2, b2, scale2, shift2);

  // 7) in-place BN2 + LeakyReLU finalize
  bn_act_kernel<<<(int)(((size_t)B_ * C2_ * N0_) / 256), 256, 0, stream>>>(
      out_nf, scale2, shift2, C2_);
}